// UP_GAT_2413771621185
// MI455X (gfx1250) — compile-verified
//
#include <hip/hip_runtime.h>
#include <math.h>

// Problem constants (match reference)
#define B_  32
#define N_  8192
#define D_  256
#define H_  4
#define KF_ 819          // int(0.1 * 8192)
#define NCHUNK_ 64
#define NSLICE_ (N_ / NCHUNK_)   // 128

typedef float v2f __attribute__((ext_vector_type(2)));
typedef float v8f __attribute__((ext_vector_type(8)));

// ---------------------------------------------------------------------------
// WMMA fp32 GEMM: C[32 x 256] = A[32 x 256] * B' + bias, using
// V_WMMA_F32_16X16X4_F32, one wave per 16x16 tile (grid 16 x 2, block 32).
// WHICH selects how B' elements are fetched:
//   0: B'(k,n) = Bm[n*256 + k]                       (q = u @ Wq^T)
//   1: B'(k,n) = Bm[k*256 + n]                       (v = q @ Wk)
//   2: B'(k,n) = sum_h Bm[n*1024 + h*256 + k]        (out = head @ Wo_sum^T)
// ---------------------------------------------------------------------------
template <int WHICH>
__global__ void wmma_gemm32(const float* __restrict__ A,
                            const float* __restrict__ Bm,
                            const float* __restrict__ bias,
                            float* __restrict__ C) {
    const int lane = threadIdx.x;            // 0..31, full wave32
    const int half = lane >> 4;              // 0 or 1
    const int l15  = lane & 15;
    const int mt   = blockIdx.y * 16;
    const int nt   = blockIdx.x * 16;
    const int mrow = mt + l15;               // A row this lane feeds
    const int ncol = nt + l15;               // B/C/D column this lane feeds

    v8f acc = {0.f, 0.f, 0.f, 0.f, 0.f, 0.f, 0.f, 0.f};

    for (int k = 0; k < D_; k += 4) {
        const int ka = k + 2 * half;         // K index of first VGPR for this lane
        v2f a;
        a.x = A[mrow * D_ + ka];             // A VGPR0: K = 2*half
        a.y = A[mrow * D_ + ka + 1];         // A VGPR1: K = 2*half + 1
        v2f b;
        if (WHICH == 0) {
            b.x = Bm[ncol * D_ + ka];
            b.y = Bm[ncol * D_ + ka + 1];
        } else if (WHICH == 1) {
            b.x = Bm[ka * D_ + ncol];
            b.y = Bm[(ka + 1) * D_ + ncol];
        } else {
            const float* r = Bm + (size_t)ncol * (D_ * H_);
            b.x = r[ka] + r[D_ + ka] + r[2 * D_ + ka] + r[3 * D_ + ka];
            b.y = r[ka + 1] + r[D_ + ka + 1] + r[2 * D_ + ka + 1] + r[3 * D_ + ka + 1];
        }
        acc = __builtin_amdgcn_wmma_f32_16x16x4_f32(
            /*neg_a=*/false, a, /*neg_b=*/false, b,
            /*c_mod=*/(short)0, acc, /*reuse_a=*/false, /*reuse_b=*/false);
    }

    const float bb = bias ? bias[ncol] : 0.0f;
#pragma unroll
    for (int j = 0; j < 8; ++j) {
        const int row = mt + j + half * 8;   // C/D: VGPR j -> rows j and j+8
        C[row * D_ + ncol] = acc[j] + bb;
    }
}

// c[b] = q[b,:] . bk  (bk is zeros in the reference, computed for fidelity)
__global__ void dot_bias_kernel(const float* __restrict__ q,
                                const float* __restrict__ bk,
                                float* __restrict__ c) {
    __shared__ float red[256];
    const int b = blockIdx.x, tid = threadIdx.x;
    red[tid] = q[b * D_ + tid] * bk[tid];
    __syncthreads();
    for (int st = 128; st > 0; st >>= 1) {
        if (tid < st) red[tid] += red[tid + st];
        __syncthreads();
    }
    if (tid == 0) c[b] = red[0];
}

// ---------------------------------------------------------------------------
// Pass 1 (bandwidth bound, 256 MB): s[b,n] = (v[b]·p[b,n] + c[b]) / 16,
// masked positions get raw score 1e-8 (faithful to reference masked_fill).
// One wave per (b,n) row; float4 loads; shfl-xor wave32 reduction.
// ---------------------------------------------------------------------------
__global__ void scores_kernel(const float* __restrict__ v,
                              const float* __restrict__ c,
                              const float* __restrict__ p,
                              const int* __restrict__ mask,
                              float* __restrict__ s) {
    const int wave = threadIdx.x >> 5;
    const int lane = threadIdx.x & 31;
    const long long row = (long long)blockIdx.x * 8 + wave;  // 0 .. B*N-1
    const int b = (int)(row >> 13);                          // / 8192

    const float4* prow = (const float4*)(p + (row << 8));    // row * 256 floats
    const float4* vb   = (const float4*)(v + b * D_);

    const float4 p0 = prow[lane * 2 + 0];
    const float4 p1 = prow[lane * 2 + 1];
    const float4 v0 = vb[lane * 2 + 0];
    const float4 v1 = vb[lane * 2 + 1];

    float acc = p0.x * v0.x + p0.y * v0.y + p0.z * v0.z + p0.w * v0.w
              + p1.x * v1.x + p1.y * v1.y + p1.z * v1.z + p1.w * v1.w;
    for (int off = 16; off > 0; off >>= 1)
        acc += __shfl_xor(acc, off, 32);

    if (lane == 0) {
        float sc = (acc + c[b]) * 0.0625f;   // 1/sqrt(256)
        if (mask[row] == 0) sc = 1e-8f;
        s[row] = sc;
    }
}

// order-preserving float <-> uint key mapping
__device__ __forceinline__ unsigned floatToKey(float f) {
    unsigned u = __float_as_uint(f);
    return (u & 0x80000000u) ? ~u : (u | 0x80000000u);
}
__device__ __forceinline__ float keyToFloat(unsigned u) {
    unsigned bits = (u & 0x80000000u) ? (u ^ 0x80000000u) : ~u;
    return __uint_as_float(bits);
}

// ---------------------------------------------------------------------------
// Per-batch softmax stats + 819th-smallest selection (bisection on keys in
// LDS) + weight emission. softmax is strictly monotone in the scores, so
// rank(attn) == rank(score): select on keys, compare attn directly.
// ---------------------------------------------------------------------------
__global__ void softmax_filter_kernel(const float* __restrict__ s,
                                      float* __restrict__ w) {
    __shared__ unsigned keys[N_];     // 32 KB of the 320 KB WGP LDS
    __shared__ float    redf[256];
    __shared__ int      redi[256];

    const int b = blockIdx.x, tid = threadIdx.x;
    const float* sb = s + (size_t)b * N_;

    float m = -3.4e38f;
    for (int n = tid; n < N_; n += 256) {
        const float f = sb[n];
        keys[n] = floatToKey(f);
        m = fmaxf(m, f);
    }
    redf[tid] = m;
    __syncthreads();
    for (int st = 128; st > 0; st >>= 1) {
        if (tid < st) redf[tid] = fmaxf(redf[tid], redf[tid + st]);
        __syncthreads();
    }
    m = redf[0];
    __syncthreads();

    float z = 0.f;
    for (int n = tid; n < N_; n += 256)
        z += expf(keyToFloat(keys[n]) - m);
    redf[tid] = z;
    __syncthreads();
    for (int st = 128; st > 0; st >>= 1) {
        if (tid < st) redf[tid] += redf[tid + st];
        __syncthreads();
    }
    z = redf[0];
    __syncthreads();
    const float invZ = 1.0f / z;

    // find key of the KF_-th smallest element: smallest u with count(<= u) >= KF_
    unsigned lo = 0u, hi = 0xFFFFFFFFu;
    while (lo < hi) {
        const unsigned mid = lo + ((hi - lo) >> 1);
        int cnt = 0;
        for (int n = tid; n < N_; n += 256)
            cnt += (keys[n] <= mid) ? 1 : 0;
        redi[tid] = cnt;
        __syncthreads();
        for (int st = 128; st > 0; st >>= 1) {
            if (tid < st) redi[tid] += redi[tid + st];
            __syncthreads();
        }
        cnt = redi[0];
        __syncthreads();
        if (cnt >= KF_) hi = mid; else lo = mid + 1;
    }
    const unsigned thrKey = lo;

    for (int n = tid; n < N_; n += 256) {
        const unsigned u = keys[n];
        const float f = keyToFloat(u);
        const float attn = expf(f - m) * invZ;
        // (attn < thr) <=> (score < kth-smallest score) <=> (key < thrKey)
        const float wt = ((attn < 0.02f) && (u < thrKey)) ? 1e-8f : attn;
        w[(size_t)b * N_ + n] = wt;
    }
}

// ---------------------------------------------------------------------------
// Pass 2 (bandwidth bound, 256 MB): head[b,e] = sum_n w[b,n] * p[b,n,e],
// split over NCHUNK_ n-slices per batch for occupancy; deterministic.
// ---------------------------------------------------------------------------
__global__ void head_partial_kernel(const float* __restrict__ w,
                                    const float* __restrict__ p,
                                    float* __restrict__ part) {
    const int b = blockIdx.y, ch = blockIdx.x, e = threadIdx.x;  // e in [0,256)
    const long long n0 = (long long)b * N_ + (long long)ch * NSLICE_;
    const float* pb = p + (n0 << 8);
    const float* wb = w + n0;
    float acc = 0.f;
#pragma unroll 4
    for (int i = 0; i < NSLICE_; ++i)
        acc = fmaf(wb[i], pb[((long long)i << 8) + e], acc);
    part[(((size_t)b * NCHUNK_) + ch) * D_ + e] = acc;
}

__global__ void head_reduce_kernel(const float* __restrict__ part,
                                   float* __restrict__ head) {
    const int idx = blockIdx.x * 256 + threadIdx.x;   // 0 .. 8191
    const int b = idx >> 8, e = idx & 255;
    float acc = 0.f;
#pragma unroll
    for (int ch = 0; ch < NCHUNK_; ++ch)
        acc += part[(((size_t)b * NCHUNK_) + ch) * D_ + e];
    head[idx] = acc;
}

// ---------------------------------------------------------------------------
extern "C" void kernel_launch(void* const* d_in, const int* in_sizes, int n_in,
                              void* d_out, int out_size, void* d_ws, size_t ws_size,
                              hipStream_t stream) {
    (void)in_sizes; (void)n_in; (void)out_size; (void)ws_size;

    const int*   p_mask = (const int*)d_in[0];
    const float* p_feat = (const float*)d_in[1];
    const float* u_feat = (const float*)d_in[2];
    const float* Wq     = (const float*)d_in[3];
    const float* bq     = (const float*)d_in[4];
    const float* Wk     = (const float*)d_in[5];
    const float* bk     = (const float*)d_in[6];
    const float* Wo     = (const float*)d_in[7];
    const float* bo     = (const float*)d_in[8];
    float* out = (float*)d_out;

    char* ws = (char*)d_ws;
    size_t o = 0;
    float* q      = (float*)(ws + o); o += (size_t)B_ * D_ * 4;            // 32 KB
    float* v      = (float*)(ws + o); o += (size_t)B_ * D_ * 4;            // 32 KB
    float* cvec   = (float*)(ws + o); o += 256 * 4;                        // pad
    float* scores = (float*)(ws + o); o += (size_t)B_ * N_ * 4;            // 1 MB
    float* wbuf   = (float*)(ws + o); o += (size_t)B_ * N_ * 4;            // 1 MB
    float* part   = (float*)(ws + o); o += (size_t)B_ * NCHUNK_ * D_ * 4;  // 2 MB
    float* head   = (float*)(ws + o); o += (size_t)B_ * D_ * 4;            // 32 KB

    const dim3 gTiles(D_ / 16, B_ / 16);   // 16 x 2 tiles, one wave each

    // q = u @ Wq^T + bq                                  [32 x 256]
    wmma_gemm32<0><<<gTiles, 32, 0, stream>>>(u_feat, Wq, bq, q);
    // v = q @ Wk   (Wk folded into the query side)       [32 x 256]
    wmma_gemm32<1><<<gTiles, 32, 0, stream>>>(q, Wk, nullptr, v);
    // c[b] = q[b] . bk
    dot_bias_kernel<<<B_, 256, 0, stream>>>(q, bk, cvec);
    // scores[b,n] = (v[b].p[b,n] + c[b]) / 16, masked -> 1e-8
    scores_kernel<<<(B_ * N_) / 8, 256, 0, stream>>>(v, cvec, p_feat, p_mask, scores);
    // softmax + kth-smallest threshold -> weights
    softmax_filter_kernel<<<B_, 256, 0, stream>>>(scores, wbuf);
    // head[b,e] = sum_n w[b,n] p[b,n,e]
    head_partial_kernel<<<dim3(NCHUNK_, B_), 256, 0, stream>>>(wbuf, p_feat, part);
    head_reduce_kernel<<<B_, 256, 0, stream>>>(part, head);
    // out = head @ Wo_sum^T + bo   (tile(head,H) @ Wo^T == head @ sum_h Wo_h^T)
    wmma_gemm32<2><<<gTiles, 32, 0, stream>>>(head, Wo, bo, out);
}